// GatedAttentionWithPairBias_82806969466886
// MI455X (gfx1250) — compile-verified
//
#include <hip/hip_runtime.h>

typedef float v2f __attribute__((ext_vector_type(2)));
typedef float v4f __attribute__((ext_vector_type(4)));
typedef float v8f __attribute__((ext_vector_type(8)));

#define DEV static __device__ __forceinline__

constexpr int N = 768;
constexpr int NODE_DIM = 384;
constexpr int EDGE_DIM = 128;
constexpr int H = 8;
constexpr int D = 32;
constexpr int INNER = 256;      // H*D
constexpr int QKV_COLS = 768;   // 3*INNER

// ---------- wave32 reductions ----------
DEV float wave_sum(float v) {
#pragma unroll
  for (int m = 16; m >= 1; m >>= 1) v += __shfl_xor(v, m, 32);
  return v;
}
DEV float wave_max(float v) {
#pragma unroll
  for (int m = 16; m >= 1; m >>= 1) v = fmaxf(v, __shfl_xor(v, m, 32));
  return v;
}

// ---------- fp32 WMMA 16x16x4 (D = A(16x4) * B(4x16) + C) ----------
DEV v8f wmma_f32(v2f a, v2f b, v8f c) {
  return __builtin_amdgcn_wmma_f32_16x16x4_f32(
      /*neg_a=*/false, a, /*neg_b=*/false, b,
      /*c_mod=*/(short)0, c, /*reuse_a=*/false, /*reuse_b=*/false);
}

// =====================================================================
// 1) LayerNorm of node features: one wave per row (384 cols, 12/lane)
// =====================================================================
__global__ void k_ln_nodes(const float* __restrict__ x,
                           const float* __restrict__ g,
                           const float* __restrict__ b,
                           float* __restrict__ y) {
  const int lane = threadIdx.x & 31;
  const int row  = blockIdx.x * 8 + (threadIdx.x >> 5);
  const float* p = x + (size_t)row * NODE_DIM;
  float vals[12];
  float s = 0.f, ss = 0.f;
#pragma unroll
  for (int t = 0; t < 12; ++t) {
    float v = p[lane + 32 * t];
    vals[t] = v; s += v; ss += v * v;
  }
  s  = wave_sum(s);
  ss = wave_sum(ss);
  const float mean = s * (1.0f / NODE_DIM);
  const float var  = ss * (1.0f / NODE_DIM) - mean * mean;
  const float rstd = rsqrtf(var + 1e-5f);
  float* o = y + (size_t)row * NODE_DIM;
#pragma unroll
  for (int t = 0; t < 12; ++t) {
    const int c = lane + 32 * t;
    o[c] = (vals[t] - mean) * rstd * g[c] + b[c];
  }
}

// =====================================================================
// 2) QKV GEMM (split-K x8 in workgroup): C[768,768] = normed @ W_qkv
//    8 waves/block, each covers K-chunk of 48; LDS tree-reduce.
// =====================================================================
__global__ void k_qkv(const float* __restrict__ A,
                      const float* __restrict__ B,
                      float* __restrict__ C) {
  __shared__ float red[8 * 256];
  const int lane = threadIdx.x & 31;
  const int w    = threadIdx.x >> 5;       // k-chunk 0..7
  const int tile = blockIdx.x;             // 48*48 tiles
  const int m0 = (tile / 48) * 16, n0 = (tile % 48) * 16;
  const int half = lane >> 4, idx = lane & 15;
  const int k0 = w * 48;
  const float* ar = A + (size_t)(m0 + idx) * NODE_DIM + k0 + 2 * half;
  const float* bc = B + (size_t)(k0 + 2 * half) * QKV_COLS + n0 + idx;
  v8f acc = {};
#pragma unroll
  for (int k = 0; k < 48; k += 4) {
    v2f a = { ar[k], ar[k + 1] };
    v2f b = { bc[(size_t)k * QKV_COLS], bc[(size_t)(k + 1) * QKV_COLS] };
    acc = wmma_f32(a, b, acc);
  }
  float* rw = red + w * 256;
#pragma unroll
  for (int v = 0; v < 8; ++v) rw[(8 * half + v) * 16 + idx] = acc[v];
  __syncthreads();
  const int t = threadIdx.x;
  float s = 0.f;
#pragma unroll
  for (int u = 0; u < 8; ++u) s += red[u * 256 + t];
  C[(size_t)(m0 + (t >> 4)) * QKV_COLS + n0 + (t & 15)] = s;
}

// =====================================================================
// 3) Gate GEMM (split-K x8): G = sigmoid(node @ W_gate + b_gate)
// =====================================================================
__global__ void k_gate(const float* __restrict__ A,
                       const float* __restrict__ B,
                       const float* __restrict__ bg,
                       float* __restrict__ G) {
  __shared__ float red[8 * 256];
  const int lane = threadIdx.x & 31;
  const int w    = threadIdx.x >> 5;
  const int tile = blockIdx.x;             // 48*16 tiles
  const int m0 = (tile / 16) * 16, n0 = (tile % 16) * 16;
  const int half = lane >> 4, idx = lane & 15;
  const int k0 = w * 48;
  const float* ar = A + (size_t)(m0 + idx) * NODE_DIM + k0 + 2 * half;
  const float* bc = B + (size_t)(k0 + 2 * half) * INNER + n0 + idx;
  v8f acc = {};
#pragma unroll
  for (int k = 0; k < 48; k += 4) {
    v2f a = { ar[k], ar[k + 1] };
    v2f b = { bc[(size_t)k * INNER], bc[(size_t)(k + 1) * INNER] };
    acc = wmma_f32(a, b, acc);
  }
  float* rw = red + w * 256;
#pragma unroll
  for (int v = 0; v < 8; ++v) rw[(8 * half + v) * 16 + idx] = acc[v];
  __syncthreads();
  const int t = threadIdx.x;
  float s = 0.f;
#pragma unroll
  for (int u = 0; u < 8; ++u) s += red[u * 256 + t];
  const int col = n0 + (t & 15);
  const float z = s + bg[col];
  G[(size_t)(m0 + (t >> 4)) * INNER + col] = 1.0f / (1.0f + __expf(-z));
}

// =====================================================================
// 4a) Prep: fold LN affine into the pair projection.
// =====================================================================
__global__ void k_pair_prep(const float* __restrict__ g,
                            const float* __restrict__ b,
                            const float* __restrict__ Wp,
                            float* __restrict__ A,
                            float* __restrict__ SA,
                            float* __restrict__ Cb) {
  const int t = threadIdx.x;  // 256 threads
  for (int i = t; i < EDGE_DIM * H; i += 256) A[i] = g[i >> 3] * Wp[i];
  if (t < H) {
    float sa = 0.f, cb = 0.f;
    for (int c = 0; c < EDGE_DIM; ++c) {
      sa += g[c] * Wp[c * H + t];
      cb += b[c] * Wp[c * H + t];
    }
    SA[t] = sa; Cb[t] = cb;
  }
}

// =====================================================================
// 4b) Pair bias: 302MB streaming pass, one position per thread.
//     bias[h] = rstd*(dot(x,A[:,h]) - mean*SA[h]) + Cb[h]
// =====================================================================
__global__ void k_pair(const float* __restrict__ edge,
                       const float* __restrict__ A,
                       const float* __restrict__ SA,
                       const float* __restrict__ Cb,
                       float* __restrict__ bias) {
  const size_t pos = (size_t)blockIdx.x * 256 + threadIdx.x;  // i*N + j
  const v4f* e = (const v4f*)(edge + pos * EDGE_DIM);
  v4f x[8];
#pragma unroll
  for (int t = 0; t < 8; ++t)
    x[t] = __builtin_nontemporal_load(e + t);  // streaming: keep L2 for hot data

  float s = 0.f, ss = 0.f;
#pragma unroll
  for (int t = 0; t < 8; ++t) {
    s  += x[t].x + x[t].y + x[t].z + x[t].w;
    ss += x[t].x * x[t].x + x[t].y * x[t].y + x[t].z * x[t].z + x[t].w * x[t].w;
  }
  const float mean = s * (1.0f / EDGE_DIM);
  const float var  = ss * (1.0f / EDGE_DIM) - mean * mean;
  const float rstd = rsqrtf(var + 1e-5f);

  float acc[8] = {};
#pragma unroll
  for (int t = 0; t < 8; ++t) {
    const float xs[4] = { x[t].x, x[t].y, x[t].z, x[t].w };
#pragma unroll
    for (int u = 0; u < 4; ++u) {
      const float xv = xs[u];
      const float* aw = A + (size_t)(4 * t + u) * H;  // uniform -> SGPRs
#pragma unroll
      for (int h = 0; h < H; ++h) acc[h] += xv * aw[h];
    }
  }
#pragma unroll
  for (int h = 0; h < H; ++h)
    bias[(size_t)h * N * N + pos] = rstd * (acc[h] - mean * SA[h]) + Cb[h];
}

// =====================================================================
// 5) Scores: S[h,i,j] = (q_h[i]·k_h[j]) / sqrt(D) + bias[h,i,j]
//    wave per tile (18432 waves, K=32 -> already latency-friendly)
// =====================================================================
__global__ void k_scores(const float* __restrict__ qkv,
                         const float* __restrict__ bias,
                         float* __restrict__ S) {
  const int lane = threadIdx.x & 31;
  const int wid  = blockIdx.x * 8 + (threadIdx.x >> 5);
  const int h  = wid / (48 * 48);
  const int r  = wid % (48 * 48);
  const int m0 = (r / 48) * 16, n0 = (r % 48) * 16;
  const int half = lane >> 4, idx = lane & 15;
  const float* qr = qkv + (size_t)(m0 + idx) * QKV_COLS + h * D + 2 * half;
  const float* kr = qkv + (size_t)(n0 + idx) * QKV_COLS + INNER + h * D + 2 * half;
  v8f acc = {};
#pragma unroll
  for (int k = 0; k < D; k += 4) {
    v2f a = { qr[k], qr[k + 1] };
    v2f b = { kr[k], kr[k + 1] };   // B[k][n] = K[n][k]
    acc = wmma_f32(a, b, acc);
  }
  const float scale = 0.17677669529663687f; // 1/sqrt(32)
  const float* bp = bias + ((size_t)h * N + m0) * N + n0 + idx;
  float*       sp = S    + ((size_t)h * N + m0) * N + n0 + idx;
#pragma unroll
  for (int v = 0; v < 8; ++v) {
    const size_t m = 8 * half + v;
    sp[m * N] = acc[v] * scale + bp[m * N];
  }
}

// =====================================================================
// 6) Softmax over j: one wave per (h,i) row of 768 (24 elems/lane)
// =====================================================================
__global__ void k_softmax(float* __restrict__ S) {
  const int lane = threadIdx.x & 31;
  const int row  = blockIdx.x * 8 + (threadIdx.x >> 5); // 0..6143
  float* p = S + (size_t)row * N;
  float vals[24];
  float mx = -3.402823466e38f;
#pragma unroll
  for (int t = 0; t < 24; ++t) { vals[t] = p[lane + 32 * t]; mx = fmaxf(mx, vals[t]); }
  mx = wave_max(mx);
  float sum = 0.f;
#pragma unroll
  for (int t = 0; t < 24; ++t) { vals[t] = __expf(vals[t] - mx); sum += vals[t]; }
  sum = wave_sum(sum);
  const float inv = 1.0f / sum;
#pragma unroll
  for (int t = 0; t < 24; ++t) p[lane + 32 * t] = vals[t] * inv;
}

// =====================================================================
// 7) out[h,j,d] = sum_i attn[h,i,j] * v[h,i,d] (split-K x8), * gate
//    A-fragment reads attn column-major (transposed contraction).
// =====================================================================
__global__ void k_out(const float* __restrict__ S,
                      const float* __restrict__ qkv,
                      const float* __restrict__ G,
                      float* __restrict__ oin) {
  __shared__ float red[8 * 256];
  const int lane = threadIdx.x & 31;
  const int w    = threadIdx.x >> 5;
  const int tile = blockIdx.x;            // h(8) x jt(48) x dt(2)
  const int h  = tile / (48 * 2);
  const int r  = tile % (48 * 2);
  const int m0 = (r / 2) * 16;            // j tile
  const int n0 = (r % 2) * 16;            // d tile
  const int half = lane >> 4, idx = lane & 15;
  const int k0 = w * 96;                  // i-chunk
  // A[m][k] = attn[h][k][m0+idx]
  const float* ab = S + (size_t)h * N * N + (size_t)(k0 + 2 * half) * N + m0 + idx;
  // B[k][n] = v[h][k][n] = qkv[k][512 + h*32 + n]
  const float* vb = qkv + (size_t)(k0 + 2 * half) * QKV_COLS + 2 * INNER + h * D + n0 + idx;
  v8f acc = {};
#pragma unroll
  for (int k = 0; k < 96; k += 4) {
    v2f a = { ab[(size_t)k * N],        ab[(size_t)(k + 1) * N] };
    v2f b = { vb[(size_t)k * QKV_COLS], vb[(size_t)(k + 1) * QKV_COLS] };
    acc = wmma_f32(a, b, acc);
  }
  float* rw = red + w * 256;
#pragma unroll
  for (int v = 0; v < 8; ++v) rw[(8 * half + v) * 16 + idx] = acc[v];
  __syncthreads();
  const int t = threadIdx.x;
  float s = 0.f;
#pragma unroll
  for (int u = 0; u < 8; ++u) s += red[u * 256 + t];
  const int j   = m0 + (t >> 4);
  const int col = h * D + n0 + (t & 15);
  oin[(size_t)j * INNER + col] = s * G[(size_t)j * INNER + col];
}

// =====================================================================
// 8) Final (split-K x8): out = node + oin @ W_final + b_final
// =====================================================================
__global__ void k_final(const float* __restrict__ A,
                        const float* __restrict__ B,
                        const float* __restrict__ node,
                        const float* __restrict__ bf,
                        float* __restrict__ C) {
  __shared__ float red[8 * 256];
  const int lane = threadIdx.x & 31;
  const int w    = threadIdx.x >> 5;
  const int tile = blockIdx.x;            // 48 x 24 tiles
  const int m0 = (tile / 24) * 16, n0 = (tile % 24) * 16;
  const int half = lane >> 4, idx = lane & 15;
  const int k0 = w * 32;
  const float* ar = A + (size_t)(m0 + idx) * INNER + k0 + 2 * half;
  const float* bc = B + (size_t)(k0 + 2 * half) * NODE_DIM + n0 + idx;
  v8f acc = {};
#pragma unroll
  for (int k = 0; k < 32; k += 4) {
    v2f a = { ar[k], ar[k + 1] };
    v2f b = { bc[(size_t)k * NODE_DIM], bc[(size_t)(k + 1) * NODE_DIM] };
    acc = wmma_f32(a, b, acc);
  }
  float* rw = red + w * 256;
#pragma unroll
  for (int v = 0; v < 8; ++v) rw[(8 * half + v) * 16 + idx] = acc[v];
  __syncthreads();
  const int t = threadIdx.x;
  float s = 0.f;
#pragma unroll
  for (int u = 0; u < 8; ++u) s += red[u * 256 + t];
  const size_t row = m0 + (t >> 4);
  const int col = n0 + (t & 15);
  C[row * NODE_DIM + col] = s + node[row * NODE_DIM + col] + bf[col];
}

// =====================================================================
extern "C" void kernel_launch(void* const* d_in, const int* in_sizes, int n_in,
                              void* d_out, int out_size, void* d_ws, size_t ws_size,
                              hipStream_t stream) {
  const float* node   = (const float*)d_in[0];
  const float* edge   = (const float*)d_in[1];
  const float* ln1_g  = (const float*)d_in[2];
  const float* ln1_b  = (const float*)d_in[3];
  const float* W_qkv  = (const float*)d_in[4];
  const float* ln2_g  = (const float*)d_in[5];
  const float* ln2_b  = (const float*)d_in[6];
  const float* W_pair = (const float*)d_in[7];
  const float* W_gate = (const float*)d_in[8];
  const float* b_gate = (const float*)d_in[9];
  const float* W_fin  = (const float*)d_in[10];
  const float* b_fin  = (const float*)d_in[11];
  float* out = (float*)d_out;

  float* ws     = (float*)d_ws;
  float* normed = ws;                       // 768*384   = 294912
  float* qkv    = ws + 294912;              // 768*768   = 589824
  float* gate   = ws + 884736;              // 768*256   = 196608
  float* bias   = ws + 1081344;             // 8*768*768 = 4718592
  float* attn   = ws + 5799936;             // 8*768*768 = 4718592
  float* oin    = ws + 10518528;            // 768*256   = 196608
  float* pA     = ws + 10715136;            // 128*8     = 1024
  float* pSA    = ws + 10716160;            // 8
  float* pCb    = ws + 10716168;            // 8   (total ~42.9 MB)

  const dim3 blk(256);
  k_pair_prep<<<1, blk, 0, stream>>>(ln2_g, ln2_b, W_pair, pA, pSA, pCb);
  k_ln_nodes<<<N / 8, blk, 0, stream>>>(node, ln1_g, ln1_b, normed);
  k_qkv<<<48 * 48, blk, 0, stream>>>(normed, W_qkv, qkv);
  k_gate<<<48 * 16, blk, 0, stream>>>(node, W_gate, b_gate, gate);
  k_pair<<<(N * N) / 256, blk, 0, stream>>>(edge, pA, pSA, pCb, bias);
  k_scores<<<(H * 48 * 48) / 8, blk, 0, stream>>>(qkv, bias, attn);
  k_softmax<<<(H * N) / 8, blk, 0, stream>>>(attn);
  k_out<<<H * 48 * 2, blk, 0, stream>>>(attn, qkv, gate, oin);
  k_final<<<48 * 24, blk, 0, stream>>>(oin, W_fin, node, b_fin, out);
}